// MessagePassingLayer_188978561157
// MI455X (gfx1250) — compile-verified
//
#include <hip/hip_runtime.h>

#define N_ENT 50000
#define N_RELS 500
#define DD 128
#define NE 600000

typedef __attribute__((ext_vector_type(2))) float v2f;
typedef __attribute__((ext_vector_type(8))) float v8f;

// ---------------- degree counting ----------------
__global__ void deg_kernel(const int* __restrict__ ei,
                           unsigned int* __restrict__ deg_r,
                           unsigned int* __restrict__ deg_c) {
    int e = blockIdx.x * blockDim.x + threadIdx.x;
    if (e < NE) {
        atomicAdd(&deg_r[ei[e]], 1u);        // row ids
        atomicAdd(&deg_c[ei[NE + e]], 1u);   // col ids
    }
}

// convert counts -> deg^{-1/2} in place (reinterpret as float)
__global__ void dinv_kernel(void* dr_, void* dc_) {
    int i = blockIdx.x * blockDim.x + threadIdx.x;
    if (i < N_ENT) {
        unsigned int a = ((unsigned int*)dr_)[i];
        unsigned int b = ((unsigned int*)dc_)[i];
        ((float*)dr_)[i] = (a > 0u) ? (1.0f / sqrtf((float)a)) : 0.0f;
        ((float*)dc_)[i] = (b > 0u) ? (1.0f / sqrtf((float)b)) : 0.0f;
    }
}

// ---------------- per-edge composed aggregation (scatter-add) ----------------
// One wave (32 lanes) per edge; each lane owns 4 contiguous features (float4).
// agg_in[row]  += n_in  * (x[col] - r[rel])
// agg_out[col] += n_out * (x[row] - r[rel+N_RELS])
__global__ __launch_bounds__(256) void edge_agg_kernel(
        const float* __restrict__ x, const float* __restrict__ r,
        const int* __restrict__ ei, const int* __restrict__ et,
        const float* __restrict__ dinv_r, const float* __restrict__ dinv_c,
        float* __restrict__ agg_in, float* __restrict__ agg_out) {
    int lane = threadIdx.x & 31;
    int e = blockIdx.x * 8 + (threadIdx.x >> 5);   // 8 waves/block, 1 edge/wave
    int d0 = lane * 4;

    int row = ei[e];          // wave-uniform broadcast loads
    int col = ei[NE + e];
    int rel = et[e];

    float n_in  = dinv_r[row] * dinv_r[col];
    float n_out = dinv_c[col] * dinv_c[row];

    float4 xc = *(const float4*)(x + (size_t)col * DD + d0);
    float4 xr = *(const float4*)(x + (size_t)row * DD + d0);
    float4 ri = *(const float4*)(r + (size_t)rel * DD + d0);
    float4 ro = *(const float4*)(r + (size_t)(rel + N_RELS) * DD + d0);

    float* pi = agg_in  + (size_t)row * DD + d0;
    float* po = agg_out + (size_t)col * DD + d0;
    unsafeAtomicAdd(pi + 0, n_in * (xc.x - ri.x));
    unsafeAtomicAdd(pi + 1, n_in * (xc.y - ri.y));
    unsafeAtomicAdd(pi + 2, n_in * (xc.z - ri.z));
    unsafeAtomicAdd(pi + 3, n_in * (xc.w - ri.w));
    unsafeAtomicAdd(po + 0, n_out * (xr.x - ro.x));
    unsafeAtomicAdd(po + 1, n_out * (xr.y - ro.y));
    unsafeAtomicAdd(po + 2, n_out * (xr.z - ro.z));
    unsafeAtomicAdd(po + 3, n_out * (xr.w - ro.w));
}

// ---------------- fused output GEMM (f32 WMMA 16x16x4) ----------------
// out = (agg_in@w_in + agg_out@w_out + (x - loop_rel)@w_loop) / 3 + bias
// Each wave: 16 output cols x 4 M-tiles (64 rows/block) so each B fragment
// is loaded once and reused across 4 WMMAs.
#define MT 4
__global__ __launch_bounds__(256) void out_gemm_kernel(
        const float* __restrict__ agg_in, const float* __restrict__ agg_out,
        const float* __restrict__ x,
        const float* __restrict__ w_in, const float* __restrict__ w_out,
        const float* __restrict__ w_loop,
        const float* __restrict__ loop_rel, const float* __restrict__ bias,
        float* __restrict__ out) {
    const int lane  = threadIdx.x & 31;
    const int wv    = threadIdx.x >> 5;     // 8 waves: each owns 16 output columns
    const int n0    = wv * 16;
    const int mbase = blockIdx.x * (16 * MT);
    const int half  = lane >> 4;            // K sub-pair select
    const int l     = lane & 15;

    // Clamp A-row indices (last block may run past N_ENT); EXEC stays full.
    int mA[MT];
    #pragma unroll
    for (int t = 0; t < MT; ++t) {
        int m = mbase + t * 16 + l;
        mA[t] = (m < N_ENT) ? m : (N_ENT - 1);
    }

    v8f c[MT] = {};

    // term 1: agg_in @ w_in
    for (int k0 = 0; k0 < DD; k0 += 4) {
        int ka = k0 + 2 * half;
        v2f b;
        b.x = w_in[ka * DD + n0 + l];
        b.y = w_in[(ka + 1) * DD + n0 + l];
        #pragma unroll
        for (int t = 0; t < MT; ++t) {
            v2f a = *(const v2f*)(agg_in + (size_t)mA[t] * DD + ka);
            c[t] = __builtin_amdgcn_wmma_f32_16x16x4_f32(false, a, false, b, (short)0, c[t], false, false);
        }
    }
    // term 2: agg_out @ w_out
    for (int k0 = 0; k0 < DD; k0 += 4) {
        int ka = k0 + 2 * half;
        v2f b;
        b.x = w_out[ka * DD + n0 + l];
        b.y = w_out[(ka + 1) * DD + n0 + l];
        #pragma unroll
        for (int t = 0; t < MT; ++t) {
            v2f a = *(const v2f*)(agg_out + (size_t)mA[t] * DD + ka);
            c[t] = __builtin_amdgcn_wmma_f32_16x16x4_f32(false, a, false, b, (short)0, c[t], false, false);
        }
    }
    // term 3: (x - loop_rel) @ w_loop (compose on the fly)
    for (int k0 = 0; k0 < DD; k0 += 4) {
        int ka = k0 + 2 * half;
        float lr0 = loop_rel[ka];
        float lr1 = loop_rel[ka + 1];
        v2f b;
        b.x = w_loop[ka * DD + n0 + l];
        b.y = w_loop[(ka + 1) * DD + n0 + l];
        #pragma unroll
        for (int t = 0; t < MT; ++t) {
            v2f xa = *(const v2f*)(x + (size_t)mA[t] * DD + ka);
            v2f a;
            a.x = xa.x - lr0;
            a.y = xa.y - lr1;
            c[t] = __builtin_amdgcn_wmma_f32_16x16x4_f32(false, a, false, b, (short)0, c[t], false, false);
        }
    }

    const float s  = 1.0f / 3.0f;
    const float bv = bias[n0 + l];
    #pragma unroll
    for (int t = 0; t < MT; ++t) {
        #pragma unroll
        for (int i = 0; i < 8; ++i) {
            int rowOut = mbase + t * 16 + i + half * 8;   // C/D: M = vgpr + 8*(lane>=16)
            if (rowOut < N_ENT)
                out[(size_t)rowOut * DD + n0 + l] = c[t][i] * s + bv;
        }
    }
}

// ---------------- rel GEMM: rel = r @ w_rel  (1000 rows, guarded) ----------------
__global__ __launch_bounds__(256) void rel_gemm_kernel(
        const float* __restrict__ r, const float* __restrict__ w_rel,
        float* __restrict__ rel_out) {
    const int lane = threadIdx.x & 31;
    const int wv   = threadIdx.x >> 5;
    const int n0   = wv * 16;
    const int m0   = blockIdx.x * 16;
    const int half = lane >> 4;
    const int l    = lane & 15;
    int mA = m0 + l;
    int mClamp = (mA < 2 * N_RELS) ? mA : (2 * N_RELS - 1);  // clamp loads, EXEC stays full

    v8f c = {};
    for (int k0 = 0; k0 < DD; k0 += 4) {
        int ka = k0 + 2 * half;
        v2f a = *(const v2f*)(r + (size_t)mClamp * DD + ka);
        v2f b;
        b.x = w_rel[ka * DD + n0 + l];
        b.y = w_rel[(ka + 1) * DD + n0 + l];
        c = __builtin_amdgcn_wmma_f32_16x16x4_f32(false, a, false, b, (short)0, c, false, false);
    }
    #pragma unroll
    for (int i = 0; i < 8; ++i) {
        int rowOut = m0 + i + half * 8;
        if (rowOut < 2 * N_RELS)
            rel_out[(size_t)rowOut * DD + n0 + l] = c[i];
    }
}

extern "C" void kernel_launch(void* const* d_in, const int* in_sizes, int n_in,
                              void* d_out, int out_size, void* d_ws, size_t ws_size,
                              hipStream_t stream) {
    const float* x        = (const float*)d_in[0];
    const float* r        = (const float*)d_in[1];
    const float* w_in     = (const float*)d_in[2];
    const float* w_out    = (const float*)d_in[3];
    const float* w_loop   = (const float*)d_in[4];
    const float* w_rel    = (const float*)d_in[5];
    const float* loop_rel = (const float*)d_in[6];
    const float* bias     = (const float*)d_in[7];
    const int*   ei       = (const int*)d_in[8];
    const int*   et       = (const int*)d_in[9];

    float* out     = (float*)d_out;                       // [N_ENT, D]
    float* rel_out = out + (size_t)N_ENT * DD;            // [2*N_RELS, D]

    // workspace layout (floats): dinv_r | dinv_c | agg_in | agg_out
    float* ws      = (float*)d_ws;
    float* dinv_r  = ws;
    float* dinv_c  = ws + N_ENT;
    float* agg_in  = ws + 2 * (size_t)N_ENT;
    float* agg_out = agg_in + (size_t)N_ENT * DD;
    size_t zbytes  = (2 * (size_t)N_ENT + 2 * (size_t)N_ENT * DD) * sizeof(float);

    hipMemsetAsync(d_ws, 0, zbytes, stream);

    deg_kernel<<<(NE + 255) / 256, 256, 0, stream>>>(
        ei, (unsigned int*)dinv_r, (unsigned int*)dinv_c);

    dinv_kernel<<<(N_ENT + 255) / 256, 256, 0, stream>>>(
        (void*)dinv_r, (void*)dinv_c);

    edge_agg_kernel<<<NE / 8, 256, 0, stream>>>(
        x, r, ei, et, dinv_r, dinv_c, agg_in, agg_out);

    out_gemm_kernel<<<(N_ENT / 16 + MT - 1) / MT, 256, 0, stream>>>(
        agg_in, agg_out, x, w_in, w_out, w_loop, loop_rel, bias, out);

    rel_gemm_kernel<<<(2 * N_RELS + 15) / 16, 256, 0, stream>>>(
        r, w_rel, rel_out);
}